// GraphConvFlockingModel_75943611728684
// MI455X (gfx1250) — compile-verified
//
#include <hip/hip_runtime.h>
#include <hip/hip_bf16.h>

typedef __attribute__((ext_vector_type(2))) float v2f;
typedef __attribute__((ext_vector_type(8))) float v8f;

#define N_NODES_C 100000
#define FEAT_C    64
#define OUT_C     2

// ---------------------------------------------------------------------------
// Kernel 1: build composite matrices (tiny, one block).
//   Mc[64][16]: cols 0-1 = W_root @ W_pred, cols 2-3 = W_rel @ W_pred, rest 0
//   bc[2]     = b_rel @ W_pred + b_pred
// ---------------------------------------------------------------------------
__global__ void gc_prep(const float* __restrict__ W_rel,
                        const float* __restrict__ b_rel,
                        const float* __restrict__ W_root,
                        const float* __restrict__ W_pred,
                        const float* __restrict__ b_pred,
                        float* __restrict__ Mc,
                        float* __restrict__ bc) {
    int tid = threadIdx.x;                 // 256 threads
    for (int i = tid; i < 64 * 16; i += blockDim.x) Mc[i] = 0.0f;
    __syncthreads();
    int k = tid >> 2;                      // 0..63
    int c = tid & 3;                       // 0..3
    const float* W = (c < 2) ? W_root : W_rel;
    int cc = c & 1;
    float s = 0.0f;
    #pragma unroll 8
    for (int j = 0; j < 64; ++j) s += W[k * 64 + j] * W_pred[j * 2 + cc];
    Mc[k * 16 + c] = s;
    if (tid < 2) {
        float s2 = b_pred[tid];
        for (int j = 0; j < 64; ++j) s2 += b_rel[j] * W_pred[j * 2 + tid];
        bc[tid] = s2;
    }
}

// ---------------------------------------------------------------------------
// Kernel 2: per-node GEMM via V_WMMA_F32_16X16X4_F32.
// Each wave handles 16-node tiles: D[16x16] = h_tile[16x64] @ Mc[64x16],
// accumulated over 16 K=4 steps.  Cols 0-1 -> out (+bias), cols 2-3 -> m.
// Epilogue is branchless (pointer/bias selects) so the divergent region
// around the WMMA loop stays minimal.
//
// Fragment layouts (ISA 7.12.2):
//   A 16x4 f32 : lane L<16 holds row L     {v0=K0, v1=K1}
//                lane L>=16 holds row L-16 {v0=K2, v1=K3}
//   B 4x16 f32 : lane L<16 holds col L     {v0=K0 row, v1=K1 row}
//                lane L>=16 holds col L-16 {v0=K2 row, v1=K3 row}
//   D 16x16 f32: vgpr r, lane L -> D[r + 8*(L>=16)][L&15]
// ---------------------------------------------------------------------------
__global__ void gc_node_wmma(const float* __restrict__ pos,   // [N,32]
                             const float* __restrict__ vel,   // [N,32]
                             const float* __restrict__ Mc,    // [64][16]
                             const float* __restrict__ bc,    // [2]
                             float* __restrict__ out,         // [N,2]
                             float* __restrict__ m) {         // [N,2]
    const int lane  = threadIdx.x & 31;
    const int half  = lane >> 4;          // 0 or 1
    const int l15   = lane & 15;
    const int wIdx  = (blockIdx.x * (blockDim.x >> 5)) + (threadIdx.x >> 5);
    const int nWv   = gridDim.x * (blockDim.x >> 5);

    // Preload all 16 B fragments (Mc is 4KB, L2 resident).
    v2f Bf[16];
    #pragma unroll
    for (int t = 0; t < 16; ++t) {
        int k0 = 4 * t + 2 * half;
        v2f b;
        b.x = Mc[k0 * 16 + l15];
        b.y = Mc[(k0 + 1) * 16 + l15];
        Bf[t] = b;
    }
    const float bias0 = bc[0];
    const float bias1 = bc[1];

    // Branchless epilogue setup (uniform over the tile loop):
    //   n = l15 (only n < 4 lanes store)
    //   n 0/1 -> out column n (with bias), n 2/3 -> m column n-2
    const int   n      = l15;
    float*      basep  = (n < 2) ? out : m;
    const int   col    = n & 1;
    const float biasn  = (n == 0) ? bias0 : ((n == 1) ? bias1 : 0.0f);

    const int nTiles = N_NODES_C / 16;    // 6250 (exact)
    for (int tile = wIdx; tile < nTiles; tile += nWv) {
        const int row = tile * 16 + l15;  // this lane's A-matrix row
        const float* prow = pos + row * 32;
        const float* vrow = vel + row * 32;

        v8f acc = {0.f, 0.f, 0.f, 0.f, 0.f, 0.f, 0.f, 0.f};
        #pragma unroll
        for (int t = 0; t < 16; ++t) {
            const int koff = 4 * t + 2 * half;   // even -> 8B aligned
            v2f a;
            if (t < 8) a = *(const v2f*)(prow + koff);
            else       a = *(const v2f*)(vrow + (koff - 32));
            acc = __builtin_amdgcn_wmma_f32_16x16x4_f32(
                /*neg_a=*/false, a, /*neg_b=*/false, Bf[t],
                /*c_mod=*/(short)0, acc, /*reuse_a=*/false, /*reuse_b=*/false);
        }

        // Lanes with n<4 scatter 8 rows each; single predicate, no branches.
        if (n < 4) {
            const int nodeBase = tile * 16 + 8 * half;
            #pragma unroll
            for (int r = 0; r < 8; ++r) {
                basep[(nodeBase + r) * 2 + col] = acc[r] + biasn;
            }
        }
    }
}

// ---------------------------------------------------------------------------
// Kernel 3: edge scatter, 4 edges per thread via int4 (global_load_b128)
// index loads.  m (800KB) and out (800KB) are L2-resident, so gathers hit
// L2 and the fp32 atomics resolve at the L2 atomic units.
// ---------------------------------------------------------------------------
__global__ void gc_edge_scatter(const int4* __restrict__ src4,
                                const int4* __restrict__ dst4,
                                const int*  __restrict__ src,
                                const int*  __restrict__ dst,
                                const float* __restrict__ m,
                                float* __restrict__ out,
                                int nQuads, int nEdges) {
    const int q = blockIdx.x * blockDim.x + threadIdx.x;
    if (q < nQuads) {
        __builtin_prefetch(src4 + q + 2048, 0, 1);
        __builtin_prefetch(dst4 + q + 2048, 0, 1);
        const int4 s = src4[q];
        const int4 d = dst4[q];
        const v2f m0 = *(const v2f*)(m + s.x * 2);
        const v2f m1 = *(const v2f*)(m + s.y * 2);
        const v2f m2 = *(const v2f*)(m + s.z * 2);
        const v2f m3 = *(const v2f*)(m + s.w * 2);
        atomicAdd(&out[d.x * 2 + 0], m0.x);
        atomicAdd(&out[d.x * 2 + 1], m0.y);
        atomicAdd(&out[d.y * 2 + 0], m1.x);
        atomicAdd(&out[d.y * 2 + 1], m1.y);
        atomicAdd(&out[d.z * 2 + 0], m2.x);
        atomicAdd(&out[d.z * 2 + 1], m2.y);
        atomicAdd(&out[d.w * 2 + 0], m3.x);
        atomicAdd(&out[d.w * 2 + 1], m3.y);
    }
    // Remainder edges (nEdges not divisible by 4): handled by thread 0.
    if (q == 0) {
        for (int e = nQuads * 4; e < nEdges; ++e) {
            const int s = src[e];
            const int d = dst[e];
            const v2f mv = *(const v2f*)(m + s * 2);
            atomicAdd(&out[d * 2 + 0], mv.x);
            atomicAdd(&out[d * 2 + 1], mv.y);
        }
    }
}

// ---------------------------------------------------------------------------
// Inputs (setup_inputs order): pos, vel, edge_index, W_rel, b_rel, W_root,
// W_pred, b_pred.  Output: [N,2] fp32.
// ---------------------------------------------------------------------------
extern "C" void kernel_launch(void* const* d_in, const int* in_sizes, int n_in,
                              void* d_out, int out_size, void* d_ws, size_t ws_size,
                              hipStream_t stream) {
    const float* pos    = (const float*)d_in[0];
    const float* vel    = (const float*)d_in[1];
    const int*   edge   = (const int*)d_in[2];
    const float* W_rel  = (const float*)d_in[3];
    const float* b_rel  = (const float*)d_in[4];
    const float* W_root = (const float*)d_in[5];
    const float* W_pred = (const float*)d_in[6];
    const float* b_pred = (const float*)d_in[7];
    float* out = (float*)d_out;

    float* ws = (float*)d_ws;
    float* Mc = ws;               // 1024 floats
    float* bc = ws + 1024;        // 2 floats
    float* m  = ws + 2048;        // 200000 floats (byte offset 8192, aligned)

    const int nEdges = in_sizes[2] / 2;
    const int nQuads = nEdges / 4;

    gc_prep<<<1, 256, 0, stream>>>(W_rel, b_rel, W_root, W_pred, b_pred, Mc, bc);

    // 6250 tiles of 16 nodes; 782 blocks x 8 waves covers them once.
    gc_node_wmma<<<782, 256, 0, stream>>>(pos, vel, Mc, bc, out, m);

    const int* src = edge;
    const int* dst = edge + nEdges;
    gc_edge_scatter<<<(nQuads + 255) / 256, 256, 0, stream>>>(
        (const int4*)src, (const int4*)dst, src, dst, m, out, nQuads, nEdges);
}